// ScaledCrossEntropyLoss_884763263124
// MI455X (gfx1250) — compile-verified
//
#include <hip/hip_runtime.h>
#include <hip/hip_bf16.h>

// ---------------- problem constants ----------------
#define NSCALE 10
#define BROWS  512
#define CCOLS  32000
#define C4     (CCOLS / 4)     // 8000 float4 per row (exactly divisible)
#define TPB    512
#define NWAVE  (TPB / 32)      // 16 waves (wave32)

#define L2E 1.4426950408889634f
#define LN2 0.6931471805599453f

typedef float v8f __attribute__((ext_vector_type(8)));
typedef float v2f __attribute__((ext_vector_type(2)));
typedef int   v4i __attribute__((ext_vector_type(4)));

// pointer types matching the gfx1250 async-load builtin signature:
// (v4i addrspace(1)*, v4i addrspace(3)*, imm offset, imm cpol)
typedef __attribute__((address_space(1))) v4i* g_v4i_p;
typedef __attribute__((address_space(3))) v4i* l_v4i_p;

// ---------------- CDNA5 helpers ----------------
__device__ __forceinline__ float fast_exp2(float x) {
#if __has_builtin(__builtin_amdgcn_exp2f)
  return __builtin_amdgcn_exp2f(x);   // v_exp_f32
#else
  return __builtin_exp2f(x);          // host/device portable fallback
#endif
}
__device__ __forceinline__ float fast_log2(float x) {
#if __has_builtin(__builtin_amdgcn_logf)
  return __builtin_amdgcn_logf(x);    // v_log_f32
#else
  return __builtin_log2f(x);          // host/device portable fallback
#endif
}

// async global -> LDS copy of 16 bytes (tracked with ASYNCcnt on gfx1250)
__device__ __forceinline__ void async_copy16(const float4* g, float4* l) {
#if __has_builtin(__builtin_amdgcn_global_load_async_to_lds_b128)
  __builtin_amdgcn_global_load_async_to_lds_b128(
      (g_v4i_p)(const void*)g, (l_v4i_p)(void*)l, 0, 0);
#else
  asm volatile("global_load_async_to_lds_b128 %0, %1, off"
               :
               : "v"((l_v4i_p)(void*)l),
                 "v"((g_v4i_p)(const void*)g)
               : "memory");
#endif
}

#if __has_builtin(__builtin_amdgcn_s_wait_asynccnt)
#define WAIT_ASYNC(n)                                \
  do {                                               \
    __builtin_amdgcn_s_wait_asynccnt(n);             \
    asm volatile("" ::: "memory");                   \
  } while (0)
#else
#define WAIT_ASYNC(n) asm volatile("s_wait_asynccnt %0" ::"i"(n) : "memory")
#endif

// ---------------- wave32 reductions ----------------
__device__ __forceinline__ float wave_max32(float v) {
#pragma unroll
  for (int m = 16; m >= 1; m >>= 1) v = fmaxf(v, __shfl_xor(v, m, 32));
  return v;
}
__device__ __forceinline__ float wave_sum32(float v) {
#pragma unroll
  for (int m = 16; m >= 1; m >>= 1) v += __shfl_xor(v, m, 32);
  return v;
}

// =====================================================================
// Kernel 1: one block per row. Pass 1: row max (HBM read, primes L2 —
// the whole 65.5 MB tensor fits in MI455X's 192 MB L2). Pass 2: stream
// the L2-hot row through an LDS double buffer with CDNA5 async loads,
// accumulating 10 per-scale exp2 sums. Writes loss[b][s].
// =====================================================================
__global__ __launch_bounds__(TPB) void sce_row_kernel(
    const float* __restrict__ x, const int* __restrict__ y,
    float* __restrict__ lossWS) {
  const int b    = blockIdx.x;
  const int tid  = threadIdx.x;
  const int lane = tid & 31;
  const int wv   = tid >> 5;

  const float4* __restrict__ row4 = (const float4*)(x + (size_t)b * CCOLS);

  __shared__ float4 stage[2][TPB];        // 16 KB double buffer
  __shared__ float  redmax[NWAVE];
  __shared__ float  red[NWAVE * NSCALE];
  __shared__ float  bcast;

  // ---- pass 1: row max ----
  float m = -3.402823466e+38f;
  for (int i = tid; i < C4; i += TPB) {
    float4 v = row4[i];
    m = fmaxf(m, fmaxf(fmaxf(v.x, v.y), fmaxf(v.z, v.w)));
  }
  m = wave_max32(m);
  if (lane == 0) redmax[wv] = m;
  __syncthreads();
  if (tid == 0) {
    float mm = redmax[0];
#pragma unroll
    for (int w = 1; w < NWAVE; ++w) mm = fmaxf(mm, redmax[w]);
    bcast = mm;
  }
  __syncthreads();
  m = bcast;

  // per-scale constants: c2[s] = scale_s * log2(e); scale_s = exp((s-9)*10/9)
  float c2[NSCALE];
#pragma unroll
  for (int s = 0; s < NSCALE; ++s) {
    float t  = (float)(s - (NSCALE - 1)) * (10.0f / 9.0f);
    float sc = fast_exp2(t * L2E);
    c2[s]    = sc * L2E;
  }

  float sum[NSCALE];
#pragma unroll
  for (int s = 0; s < NSCALE; ++s) sum[s] = 0.0f;

  // ---- pass 2: async-staged streaming (row is L2-resident now) ----
  const int nchunk = (C4 + TPB - 1) / TPB;  // 16 (15 full + 320-lane tail)
  async_copy16(&row4[tid], &stage[0][tid]); // prologue: chunk 0
  for (int ck = 0; ck < nchunk; ++ck) {
    if (ck + 1 < nchunk) {
      int nxt = (ck + 1) * TPB + tid;
      if (nxt >= C4) nxt = C4 - 1;  // clamp: uniform issue keeps ASYNCcnt uniform per wave
      async_copy16(&row4[nxt], &stage[(ck + 1) & 1][tid]);
      WAIT_ASYNC(1);                // chunk ck complete (in-order per wave)
    } else {
      WAIT_ASYNC(0);
    }
    int cur = ck * TPB + tid;
    if (cur < C4) {
      float4 v = stage[ck & 1][tid];
      float dx = v.x - m, dy = v.y - m, dz = v.z - m, dw = v.w - m;
#pragma unroll
      for (int s = 0; s < NSCALE; ++s) {
        sum[s] += fast_exp2(c2[s] * dx);
        sum[s] += fast_exp2(c2[s] * dy);
        sum[s] += fast_exp2(c2[s] * dz);
        sum[s] += fast_exp2(c2[s] * dw);
      }
    }
  }

  // ---- block reduction of the 10 sums ----
#pragma unroll
  for (int s = 0; s < NSCALE; ++s) {
    float v = wave_sum32(sum[s]);
    if (lane == 0) red[wv * NSCALE + s] = v;
  }
  __syncthreads();
  if (tid < NSCALE) {
    const int s  = tid;
    float tot = 0.0f;
#pragma unroll
    for (int w = 0; w < NWAVE; ++w) tot += red[w * NSCALE + s];
    float scale = c2[s] * LN2;
    float tgt   = x[(size_t)b * CCOLS + y[b]];
    // loss = scale*m + ln2*log2(sum) - scale*tgt
    lossWS[b * NSCALE + s] = scale * (m - tgt) + LN2 * fast_log2(tot);
  }
}

// =====================================================================
// Kernel 2: single wave. Column-sum of the 512x10 loss matrix via
// chained V_WMMA_F32_16X16X4_F32 (A = ones 16x4, B = 4x16 loss tile,
// C accumulate), then mean over batch and min over scales.
// B layout (32-bit B matrix, 4x16, 2 VGPRs): component j, lane l ->
// K = 2*j + (l>=16), N = l & 15. A is all-ones so its layout is moot.
// =====================================================================
__global__ __launch_bounds__(32) void sce_reduce_kernel(
    const float* __restrict__ lossWS, float* __restrict__ out) {
  const int lane  = threadIdx.x;
  const int n     = lane & 15;
  const int khalf = lane >> 4;

  v2f a;
  a.x = 1.0f;
  a.y = 1.0f;
  v8f acc = {};

#pragma unroll 4
  for (int it = 0; it < BROWS / 4; ++it) {
    int r0 = it * 4 + (0 + khalf);      // K = 0 + khalf
    int r1 = it * 4 + (2 + khalf);      // K = 2 + khalf
    v2f bm;
    bm.x = (n < NSCALE) ? lossWS[r0 * NSCALE + n] : 0.0f;
    bm.y = (n < NSCALE) ? lossWS[r1 * NSCALE + n] : 0.0f;
    acc = __builtin_amdgcn_wmma_f32_16x16x4_f32(
        false, a, false, bm, (short)0, acc, false, false);
  }

  // D row M=0: acc[0], lanes 0..15 hold column sums (N = lane)
  float mean = acc[0] * (1.0f / (float)BROWS);
  float mn = 3.402823466e+38f;
#pragma unroll
  for (int s = 0; s < NSCALE; ++s) {
    float v = __shfl(mean, s, 32);
    mn = fminf(mn, v);
  }
  if (lane == 0) out[0] = mn;
}

// =====================================================================
extern "C" void kernel_launch(void* const* d_in, const int* in_sizes, int n_in,
                              void* d_out, int out_size, void* d_ws,
                              size_t ws_size, hipStream_t stream) {
  const float* x = (const float*)d_in[0];
  const int*   y = (const int*)d_in[1];
  float* lossWS  = (float*)d_ws;   // 512*10 fp32 = 20 KB

  sce_row_kernel<<<BROWS, TPB, 0, stream>>>(x, y, lossWS);
  sce_reduce_kernel<<<1, 32, 0, stream>>>(lossWS, (float*)d_out);
}